// ActorCriticGAT_65773129171436
// MI455X (gfx1250) — compile-verified
//
#include <hip/hip_runtime.h>
#include <math.h>

// ---------------------------------------------------------------------------
// ActorCriticGAT forward for MI455X (gfx1250, wave32).
// GEMMs via V_WMMA_F32_16X16X4_F32 (fp32-exact); edge phase via coalesced
// block-per-edge scatter with f32 atomics (the true bottleneck at 23.3 TB/s).
// ---------------------------------------------------------------------------

#define NEG_SLOPE 0.2f

typedef float v2f __attribute__((ext_vector_type(2)));
typedef float v8f __attribute__((ext_vector_type(8)));

// ---------------- WMMA GEMM:  C[M,NC] = A[M,K] @ B[K,NC]  (row-major) ------
// Block = 256 threads = 8 waves arranged 2x4 -> 32 rows x 64 cols per block.
// Each wave computes one 16x16 tile with two interleaved 16x16x4 K-chains.
// K and NC are compile-time so all strides fold into immediate offsets.
template <int K, int NC>
__global__ __launch_bounds__(256) void gemm_wmma_f32(
    const float* __restrict__ A, const float* __restrict__ B,
    float* __restrict__ C, int M)
{
    const int tid  = threadIdx.x;
    const int lane = tid & 31;
    const int wave = tid >> 5;          // 0..7
    const int half = lane >> 4;         // 0/1  (K sub-pair / row-half select)
    const int l16  = lane & 15;

    const int wr = wave >> 2;           // 0..1
    const int wc = wave & 3;            // 0..3

    const int rowbase = blockIdx.x * 32 + wr * 16;
    const int colbase = blockIdx.y * 64 + wc * 16;

    // Row m of the A fragment only contributes to row m of D, and D rows >= M
    // are never stored -- so clamping the ragged rows is safe and branch-free.
    int arow = rowbase + l16;
    arow = (arow < M) ? arow : (M - 1);
    const int bcol = colbase + l16;

    const float* __restrict__ ap = A + (size_t)arow * K + 2 * half;
    const float* __restrict__ bp = B + (size_t)(2 * half) * NC + bcol;

    v8f acc0 = {};
    v8f acc1 = {};
    #pragma unroll 4
    for (int k = 0; k < K; k += 8) {
        const v2f a0 = *(const v2f*)(ap + k);        // A[arow][k+2h .. +1]
        const v2f a1 = *(const v2f*)(ap + k + 4);
        v2f b0, b1;                                   // B mirrors A (K pair/lane)
        b0.x = bp[(size_t)(k    ) * NC];
        b0.y = bp[(size_t)(k + 1) * NC];
        b1.x = bp[(size_t)(k + 4) * NC];
        b1.y = bp[(size_t)(k + 5) * NC];

        acc0 = __builtin_amdgcn_wmma_f32_16x16x4_f32(false, a0, false, b0,
                                                     (short)0, acc0, false, false);
        acc1 = __builtin_amdgcn_wmma_f32_16x16x4_f32(false, a1, false, b1,
                                                     (short)0, acc1, false, false);
    }
    const v8f acc = acc0 + acc1;

    // C/D layout: lanes 0-15 -> M = v, N = lane; lanes 16-31 -> M = v+8.
    #pragma unroll
    for (int v = 0; v < 8; ++v) {
        const int r = rowbase + v + 8 * half;
        if (r < M) C[(size_t)r * NC + colbase + l16] = acc[v];
    }
}

// ---------------- small utility kernels ------------------------------------
__global__ void fill_f32(float* __restrict__ p, float v, int n)
{
    int i = blockIdx.x * 256 + threadIdx.x;
    if (i < n) p[i] = v;
}

// al_s[n,h] = sum_c h[n,h,c]*a_src[h,c]; al_d likewise.
__global__ void attn_logits(const float* __restrict__ h,
                            const float* __restrict__ a_src,
                            const float* __restrict__ a_dst,
                            float* __restrict__ als, float* __restrict__ ald,
                            int N, int H, int C)
{
    int t = blockIdx.x * 256 + threadIdx.x;
    if (t >= N * H) return;
    const int n  = t / H;
    const int hh = t - n * H;
    const float* hp = h + (size_t)n * H * C + (size_t)hh * C;
    const float* as = a_src + hh * C;
    const float* ad = a_dst + hh * C;
    float ss = 0.f, sd = 0.f;
    #pragma unroll 8
    for (int c = 0; c < C; ++c) {
        const float v = hp[c];
        ss += v * as[c];
        sd += v * ad[c];
    }
    als[t] = ss;
    ald[t] = sd;
}

// Monotone float atomic-max via sign-aware int punning
// (lowers to global_atomic_max_i32 / global_atomic_min_u32).
__device__ inline void atomicMaxF32(float* addr, float val)
{
    if (val >= 0.f) atomicMax((int*)addr, __float_as_int(val));
    else            atomicMin((unsigned int*)addr, (unsigned int)__float_as_int(val));
}

// Pass 1: e = leaky_relu(al_s[src]+al_d[dst]); segment max over dst.
__global__ void edge_max(const int* __restrict__ src, const int* __restrict__ dst,
                         const float* __restrict__ als, const float* __restrict__ ald,
                         float* __restrict__ est, float* __restrict__ m,
                         int E, int Et, int H)
{
    int t = blockIdx.x * 256 + threadIdx.x;
    if (t >= Et * H) return;
    const int e  = t / H;
    const int hh = t - e * H;
    const int s = (e < E) ? src[e] : (e - E);   // appended self-loops
    const int d = (e < E) ? dst[e] : (e - E);
    float v = als[s * H + hh] + ald[d * H + hh];
    v = (v > 0.f) ? v : NEG_SLOPE * v;
    est[t] = v;
    atomicMaxF32(&m[d * H + hh], v);
}

// Pass 2: ex = exp(e - m[dst]); segment sum over dst.
__global__ void edge_exp(const int* __restrict__ dst,
                         float* __restrict__ est, const float* __restrict__ m,
                         float* __restrict__ den, int E, int Et, int H)
{
    int t = blockIdx.x * 256 + threadIdx.x;
    if (t >= Et * H) return;
    const int e  = t / H;
    const int hh = t - e * H;
    const int d  = (e < E) ? dst[e] : (e - E);
    const float ex = expf(est[t] - m[d * H + hh]);
    est[t] = ex;
    atomicAdd(&den[d * H + hh], ex);
}

// Pass 3: out[dst] += h[src] * alpha  (coalesced: one channel per thread,
// 256/HC edges per 256-thread block).
__global__ __launch_bounds__(256) void edge_agg(
    const int* __restrict__ src, const int* __restrict__ dst,
    const float* __restrict__ hlin, const float* __restrict__ est,
    const float* __restrict__ den, float* __restrict__ out,
    int E, int Et, int H, int C)
{
    const int hc  = H * C;
    const int epb = 256 / hc;
    const int le  = threadIdx.x / hc;
    const int ch  = threadIdx.x - le * hc;
    const int hh  = ch / C;
    const long e  = (long)blockIdx.x * epb + le;
    if (e >= Et) return;
    const int s = (e < E) ? src[e] : (int)(e - E);
    const int d = (e < E) ? dst[e] : (int)(e - E);
    const float alpha = est[e * H + hh] / (den[d * H + hh] + 1e-16f);
    atomicAdd(&out[(size_t)d * hc + ch], hlin[(size_t)s * hc + ch] * alpha);
}

// out += bias (broadcast over rows), optional ELU.
__global__ void bias_act(float* __restrict__ x, const float* __restrict__ b,
                         int n, int hc, int do_elu)
{
    int i = blockIdx.x * 256 + threadIdx.x;
    if (i >= n) return;
    float v = x[i] + b[i % hc];
    if (do_elu) v = (v > 0.f) ? v : (expf(v) - 1.f);
    x[i] = v;
}

// Partial column sums of emb[N,64] -> 64 atomics per block.
__global__ __launch_bounds__(256) void mean_reduce(const float* __restrict__ emb,
                                                   float* __restrict__ gsum, int N)
{
    __shared__ float sh[256];
    const int c  = threadIdx.x & 63;
    const int rl = threadIdx.x >> 6;   // 0..3
    float acc = 0.f;
    for (int r = blockIdx.x * 4 + rl; r < N; r += gridDim.x * 4)
        acc += emb[(size_t)r * 64 + c];
    sh[threadIdx.x] = acc;
    __syncthreads();
    if (threadIdx.x < 64) {
        const float s = sh[threadIdx.x] + sh[threadIdx.x + 64] +
                        sh[threadIdx.x + 128] + sh[threadIdx.x + 192];
        atomicAdd(&gsum[c], s);
    }
}

// Tiny policy/value head (O(400) flops -> single thread).
__global__ void final_head(const float* __restrict__ gsum,
                           const float* __restrict__ emb,
                           const int* __restrict__ idxp,
                           const float* __restrict__ Wp, const float* __restrict__ bp,
                           const float* __restrict__ Wv, const float* __restrict__ bv,
                           float* __restrict__ out, int N)
{
    if (threadIdx.x != 0 || blockIdx.x != 0) return;
    float g[64];
    const float inv = 1.f / (float)N;
    for (int c = 0; c < 64; ++c) g[c] = gsum[c] * inv;

    float sv = bv[0];
    for (int c = 0; c < 64; ++c) sv += g[c] * Wv[c];

    const int idx = idxp[0];
    float l0 = bp[0], l1 = bp[1];
    for (int c = 0; c < 64; ++c) {
        const float nv = emb[(size_t)idx * 64 + c];
        l0 += nv * Wp[c * 2 + 0];
        l1 += nv * Wp[c * 2 + 1];
    }
    for (int c = 0; c < 64; ++c) {
        l0 += g[c] * Wp[(64 + c) * 2 + 0];
        l1 += g[c] * Wp[(64 + c) * 2 + 1];
    }
    const float mx = fmaxf(l0, l1);
    const float e0 = expf(l0 - mx), e1 = expf(l1 - mx);
    const float dn = e0 + e1;
    out[0] = e0 / dn;
    out[1] = e1 / dn;
    out[2] = sv;
}

// ---------------------------------------------------------------------------
extern "C" void kernel_launch(void* const* d_in, const int* in_sizes, int n_in,
                              void* d_out, int out_size, void* d_ws, size_t ws_size,
                              hipStream_t stream)
{
    (void)n_in; (void)out_size; (void)ws_size;
    const float* x    = (const float*)d_in[0];
    const int*   esrc = (const int*)  d_in[1];
    const int*   edst = (const int*)  d_in[2];
    const int*   cni  = (const int*)  d_in[3];
    const float* W1   = (const float*)d_in[4];
    const float* a1s  = (const float*)d_in[5];
    const float* a1d  = (const float*)d_in[6];
    const float* b1   = (const float*)d_in[7];
    const float* W2   = (const float*)d_in[8];
    const float* a2s  = (const float*)d_in[9];
    const float* a2d  = (const float*)d_in[10];
    const float* b2   = (const float*)d_in[11];
    const float* W3   = (const float*)d_in[12];
    const float* a3s  = (const float*)d_in[13];
    const float* a3d  = (const float*)d_in[14];
    const float* b3   = (const float*)d_in[15];
    const float* Wp   = (const float*)d_in[16];
    const float* bp   = (const float*)d_in[17];
    const float* Wv   = (const float*)d_in[18];
    const float* bv   = (const float*)d_in[19];

    const int N  = in_sizes[0] / 128;   // 50000
    const int E  = in_sizes[1];         // 800000
    const int Et = E + N;               // + self loops

    // ---- workspace carve-up (floats), ~133 MB total ----
    float* ws = (float*)d_ws;
    size_t o = 0;
    float* hlin = ws + o; o += (size_t)N * 256;   // linear features (per layer)
    float* hagg = ws + o; o += (size_t)N * 256;   // aggregated features / next input
    float* emb  = ws + o; o += (size_t)N * 64;    // layer-3 output
    float* als  = ws + o; o += (size_t)N * 4;
    float* ald  = ws + o; o += (size_t)N * 4;
    float* mbuf = ws + o; o += (size_t)N * 4;
    float* den  = ws + o; o += (size_t)N * 4;
    float* est  = ws + o; o += (size_t)Et * 4;    // per-(edge,head) scalar
    float* gsum = ws + o; o += 64;

    auto cdiv = [](int a, int b) { return (a + b - 1) / b; };

    auto attn_phase = [&](const float* asrc, const float* adst, const float* bvec,
                          int H, int C, float* outbuf, int do_elu) {
        const int HC = H * C;
        attn_logits<<<cdiv(N * H, 256), 256, 0, stream>>>(hlin, asrc, adst, als, ald, N, H, C);
        fill_f32<<<cdiv(N * H, 256), 256, 0, stream>>>(mbuf, -INFINITY, N * H);
        fill_f32<<<cdiv(N * H, 256), 256, 0, stream>>>(den, 0.f, N * H);
        fill_f32<<<cdiv(N * HC, 256), 256, 0, stream>>>(outbuf, 0.f, N * HC);
        edge_max<<<cdiv(Et * H, 256), 256, 0, stream>>>(esrc, edst, als, ald, est, mbuf, E, Et, H);
        edge_exp<<<cdiv(Et * H, 256), 256, 0, stream>>>(edst, est, mbuf, den, E, Et, H);
        const int epb = 256 / HC;
        edge_agg<<<cdiv(Et, epb), 256, 0, stream>>>(esrc, edst, hlin, est, den, outbuf, E, Et, H, C);
        bias_act<<<cdiv(N * HC, 256), 256, 0, stream>>>(outbuf, bvec, N * HC, HC, do_elu);
    };

    // ---- GAT layer 1: x[N,128] -> hagg[N,256], ELU ----
    gemm_wmma_f32<128, 256><<<dim3((unsigned)cdiv(N, 32), 4), 256, 0, stream>>>(x, W1, hlin, N);
    attn_phase(a1s, a1d, b1, 4, 64, hagg, 1);

    // ---- GAT layer 2: hagg[N,256] -> hagg[N,256], ELU (stream-ordered: safe) ----
    gemm_wmma_f32<256, 256><<<dim3((unsigned)cdiv(N, 32), 4), 256, 0, stream>>>(hagg, W2, hlin, N);
    attn_phase(a2s, a2d, b2, 4, 64, hagg, 1);

    // ---- GAT layer 3: hagg[N,256] -> emb[N,64] ----
    gemm_wmma_f32<256, 64><<<dim3((unsigned)cdiv(N, 32), 1), 256, 0, stream>>>(hagg, W3, hlin, N);
    attn_phase(a3s, a3d, b3, 1, 64, emb, 0);

    // ---- readout head ----
    fill_f32<<<1, 64, 0, stream>>>(gsum, 0.f, 64);
    mean_reduce<<<256, 256, 0, stream>>>(emb, gsum, N);
    final_head<<<1, 1, 0, stream>>>(gsum, emb, cni, Wp, bp, Wv, bv, (float*)d_out, N);
}